// Monarch_61942018342943
// MI455X (gfx1250) — compile-verified
//
#include <hip/hip_runtime.h>

typedef float v2f __attribute__((ext_vector_type(2)));
typedef float v8f __attribute__((ext_vector_type(8)));

#define RANK    32
#define NCOMP   32
#define TRI     528
#define DIM     1024
#define NTOK    4096          // B*T
#define MT      16            // tokens per workgroup
#define NTILES  33            // 528 / 16
#define PSTR    536           // padded LDS row stride for P (floats)
#define KC      64            // K chunk staged in LDS per double-buffer step
#define NKC     (DIM / KC)    // 16 chunks
#define ASTR    68            // padded A-chunk row stride (floats): bank-conflict free
#define BOFF    (8 * 16 * DIM) // float offset between a wave's consecutive N-tiles (512KB)

// ---- CDNA5 async copy: global -> LDS, 16B per lane, tracked by ASYNCcnt ----
__device__ __forceinline__ void async_copy_b128(unsigned int lds_byte_off,
                                                const float* gaddr) {
    asm volatile("global_load_async_to_lds_b128 %0, %1, off"
                 :: "v"(lds_byte_off), "v"(gaddr) : "memory");
}

#if __has_builtin(__builtin_amdgcn_s_wait_asynccnt)
#define WAIT_ASYNC(n) __builtin_amdgcn_s_wait_asynccnt(n)
#else
#define WAIT_ASYNC(n) asm volatile("s_wait_asynccnt %0" :: "i"(n) : "memory")
#endif

// triangular fetch: U[r][j] for one token, P row = packed upper triangle (row-major triu)
__device__ __forceinline__ float tri_get(const float* Pt, int r, int j) {
    int off = (r * (65 - r)) >> 1;           // row start in packed triangle
    return (j >= r) ? Pt[off + (j - r)] : 0.0f;
}

__global__ __launch_bounds__(256)
void monarch_fused(const float* __restrict__ param,
                   const float* __restrict__ input,
                   const float* __restrict__ Wuv,
                   const float* __restrict__ Walpha,
                   const float* __restrict__ balpha,
                   float* __restrict__ out) {
    __shared__ float P[MT][PSTR];            // 16 x 536 floats = 34.3 KB
    __shared__ float Abuf[2][MT][ASTR];      // double-buffered A chunk, 8.7 KB

    const int m0   = blockIdx.x * MT;        // token tile base
    const int c    = blockIdx.y;             // component
    const int tid  = threadIdx.x;
    const int wid  = tid >> 5;               // wave id (0..7), wave32
    const int lane = tid & 31;
    const int half = lane >> 4;              // 0: lanes 0-15, 1: lanes 16-31
    const int lq   = lane & 15;

    // ================= Phase 1: P[m][n] = sum_k param[m0+m][k] * Wuv[c*TRI+n][k]
    // Each wave owns N-tiles {wid, wid+8, wid+16, wid+24}; tile 32 is a cleanup
    // pass by wave 0. A tile is DMA'd into LDS in 64-deep K chunks (async,
    // double buffered); B fragments stream straight from L2 (no intra-WG reuse).

    // per-lane slice of the A-chunk DMA: wave w moves rows 2w,2w+1 (512B)
    const int arow = 2 * wid + half;
    const int acol = lq * 4;
    const float* asrc = param + (size_t)(m0 + arow) * DIM + acol;
    const unsigned int adst[2] = {
        (unsigned int)(uintptr_t)&Abuf[0][arow][acol],
        (unsigned int)(uintptr_t)&Abuf[1][arow][acol] };

    v8f acc0 = {}, acc1 = {}, acc2 = {}, acc3 = {};
    // B row for this lane, first owned tile; k phase folded in (k0 = k + 2*half)
    const float* Bk = Wuv + (size_t)(c * TRI + wid * 16 + lq) * DIM + 2 * half;

    async_copy_b128(adst[0], asrc);                    // prologue: chunk 0
    for (int kc = 0; kc < NKC; ++kc) {
        WAIT_ASYNC(0);                                 // own slice of chunk kc done
        __syncthreads();                               // all slices visible; prev readers done
        if (kc + 1 < NKC)
            async_copy_b128(adst[(kc + 1) & 1], asrc + (kc + 1) * KC);

        const float* Ab = &Abuf[kc & 1][lq][2 * half];
        const float* Bc = Bk + kc * KC;
        #pragma unroll 4
        for (int s = 0; s < KC; s += 4) {
            v2f a  = *(const v2f*)(Ab + s);            // ds_load_b64, conflict-free
            v2f b0 = *(const v2f*)(Bc + s);
            v2f b1 = *(const v2f*)(Bc + s + (size_t)1 * BOFF);
            v2f b2 = *(const v2f*)(Bc + s + (size_t)2 * BOFF);
            v2f b3 = *(const v2f*)(Bc + s + (size_t)3 * BOFF);
            acc0 = __builtin_amdgcn_wmma_f32_16x16x4_f32(false, a, false, b0, (short)0, acc0, false, false);
            acc1 = __builtin_amdgcn_wmma_f32_16x16x4_f32(false, a, false, b1, (short)0, acc1, false, false);
            acc2 = __builtin_amdgcn_wmma_f32_16x16x4_f32(false, a, false, b2, (short)0, acc2, false, false);
            acc3 = __builtin_amdgcn_wmma_f32_16x16x4_f32(false, a, false, b3, (short)0, acc3, false, false);
        }
    }
    // store accumulators: C/D layout element v -> row (v + 8*half), col lq
    #pragma unroll
    for (int v = 0; v < 8; ++v) {
        P[v + 8 * half][(wid +  0) * 16 + lq] = acc0[v];
        P[v + 8 * half][(wid +  8) * 16 + lq] = acc1[v];
        P[v + 8 * half][(wid + 16) * 16 + lq] = acc2[v];
        P[v + 8 * half][(wid + 24) * 16 + lq] = acc3[v];
    }

    if (wid == 0) {                                    // cleanup: N-tile 32 (cols 512..527)
        v8f acc = {};
        const float* Arow = param + (size_t)(m0 + lq) * DIM;
        const float* Brow = Wuv + (size_t)(c * TRI + 32 * 16 + lq) * DIM;
        #pragma unroll 4
        for (int k = 0; k < DIM; k += 4) {
            const int k0 = k + 2 * half;
            v2f a = *(const v2f*)(Arow + k0);
            v2f b = *(const v2f*)(Brow + k0);
            acc = __builtin_amdgcn_wmma_f32_16x16x4_f32(false, a, false, b, (short)0, acc, false, false);
        }
        #pragma unroll
        for (int v = 0; v < 8; ++v)
            P[v + 8 * half][32 * 16 + lq] = acc[v];
    }
    __syncthreads();

    // ================= Phase 2: per-token epilogue (2 tokens per wave)
    for (int tt = 0; tt < 2; ++tt) {
        const int t     = wid * 2 + tt;
        const int token = m0 + t;
        const float* Pt = &P[t][0];

        // alpha = sigmoid(param[token] . Walpha[c] + b_alpha[c])  (in-wave dot)
        float pa = 0.0f;
        {
            const float* prow = param  + (size_t)token * DIM;
            const float* wrow = Walpha + (size_t)c * DIM;
            for (int k = lane; k < DIM; k += 32) pa += prow[k] * wrow[k];
            #pragma unroll
            for (int m = 16; m >= 1; m >>= 1) pa += __shfl_xor(pa, m, 32);
        }
        const float alpha = 1.0f / (1.0f + __expf(-(pa + balpha[c])));

        // G = U * U^T via 4 WMMA tiles; fragment f_I serves as A (rows I*16..)
        // and as B (cols I*16..) since B[k][n] = U[n][k] has identical lane layout.
        v8f g00 = {}, g01 = {}, g10 = {}, g11 = {};
        #pragma unroll
        for (int j0 = 0; j0 < RANK; j0 += 4) {
            const int jA = j0 + 2 * half;
            v2f f0, f1;
            f0.x = tri_get(Pt, lq,      jA);
            f0.y = tri_get(Pt, lq,      jA + 1);
            f1.x = tri_get(Pt, lq + 16, jA);
            f1.y = tri_get(Pt, lq + 16, jA + 1);
            g00 = __builtin_amdgcn_wmma_f32_16x16x4_f32(false, f0, false, f0, (short)0, g00, false, false);
            g01 = __builtin_amdgcn_wmma_f32_16x16x4_f32(false, f0, false, f1, (short)0, g01, false, false);
            g10 = __builtin_amdgcn_wmma_f32_16x16x4_f32(false, f1, false, f0, (short)0, g10, false, false);
            g11 = __builtin_amdgcn_wmma_f32_16x16x4_f32(false, f1, false, f1, (short)0, g11, false, false);
        }
        // Accumulator layout: gIK[v] = G[I*16 + v + 8*half][K*16 + lq]

        // column L2 norms -> max over columns
        float s0 = 0.0f, s1 = 0.0f;
        #pragma unroll
        for (int v = 0; v < 8; ++v) {
            s0 += g00[v] * g00[v] + g10[v] * g10[v];   // column lq
            s1 += g01[v] * g01[v] + g11[v] * g11[v];   // column 16+lq
        }
        s0 += __shfl_xor(s0, 16, 32);                  // fold row halves
        s1 += __shfl_xor(s1, 16, 32);
        float mx = fmaxf(sqrtf(s0), sqrtf(s1));
        #pragma unroll
        for (int m = 8; m >= 1; m >>= 1) mx = fmaxf(mx, __shfl_xor(mx, m, 32));
        const float scale = alpha / (mx + 1e-5f);

        // y = scale * (G @ x);  x[j] = input[token*DIM + c*32 + j]
        const size_t base = (size_t)token * DIM + (size_t)c * RANK;
        const float xa = input[base + lq];
        const float xb = input[base + 16 + lq];
        float y0[8], y1[8];
        #pragma unroll
        for (int v = 0; v < 8; ++v) {
            float a0 = g00[v] * xa + g01[v] * xb;      // rows  v + 8*half
            float a1 = g10[v] * xa + g11[v] * xb;      // rows 16 + v + 8*half
            #pragma unroll
            for (int m = 8; m >= 1; m >>= 1) {         // reduce over 16 cols in half
                a0 += __shfl_xor(a0, m, 32);
                a1 += __shfl_xor(a1, m, 32);
            }
            y0[v] = a0; y1[v] = a1;
        }
        if (lq == 0) {                                 // lanes 0 and 16 write their half's rows
            #pragma unroll
            for (int v = 0; v < 8; ++v) {
                const int r0 = 8 * half + v;
                const int r1 = 16 + 8 * half + v;
                out[base + r0] = input[base + r0] - scale * y0[v];
                out[base + r1] = input[base + r1] - scale * y1[v];
            }
        }
    }
}

extern "C" void kernel_launch(void* const* d_in, const int* in_sizes, int n_in,
                              void* d_out, int out_size, void* d_ws, size_t ws_size,
                              hipStream_t stream) {
    (void)in_sizes; (void)n_in; (void)out_size; (void)d_ws; (void)ws_size;
    const float* param  = (const float*)d_in[0];
    const float* input  = (const float*)d_in[1];
    const float* Wuv    = (const float*)d_in[2];
    const float* Walpha = (const float*)d_in[3];
    const float* balpha = (const float*)d_in[4];
    float* out = (float*)d_out;

    dim3 grid(NTOK / MT, NCOMP);   // 256 token tiles x 32 components
    monarch_fused<<<grid, 256, 0, stream>>>(param, input, Wuv, Walpha, balpha, out);
}